// S_Down_Sampling_33294586479300
// MI455X (gfx1250) — compile-verified
//
#include <hip/hip_runtime.h>

// out[b,t,m,c] = mean_{n in group m} in[b,t,n,c]
// P is the fixed 10x21 pooling matrix; computed as WMMA f32 16x16x4 GEMM.
//
// Groups: m0:{1,2} m1:{3,4} m2:{5,6} m3:{7,8} m4:{0,9}
//         m5:{10,11,12} m6:{13,14} m7:{15,16} m8:{17,18} m9:{19,20}

typedef __attribute__((ext_vector_type(2))) float v2f;
typedef __attribute__((ext_vector_type(8))) float v8f;

// P padded to 16x24: pool_w(m, n) = P[m][n] (0 outside the real 10x21).
// n is always a compile-time literal at the call sites below, so each call
// folds to (m == G) ? W : 0  -> a single v_cmp + v_cndmask.
__device__ __forceinline__ float pool_w(int m, int n) {
    if (n >= 21) return 0.0f;
    int g; float w;
    if (n == 0 || n == 9)      { g = 4;                  w = 0.5f;        }
    else if (n <= 8)           { g = (n - 1) >> 1;       w = 0.5f;        }
    else if (n <= 12)          { g = 5;                  w = 1.0f / 3.0f; }
    else                       { g = 6 + ((n - 13) >> 1); w = 0.5f;       }
    return (m == g) ? w : 0.0f;
}

__global__ __launch_bounds__(256)
void S_Down_Sampling_pool_wmma(const float* __restrict__ in,
                               float* __restrict__ out,
                               int n_units) {
    const int tid  = threadIdx.x;
    const int lane = tid & 31;
    const int col  = lane & 15;   // N index inside 16-wide tile / M row for A
    const int half = lane >> 4;   // which lane half (selects K pair / +8 rows)

    // one wave per unit; 8 waves per 256-thread block
    const int unit = blockIdx.x * (blockDim.x >> 5) + (tid >> 5);
    if (unit >= n_units) return;  // wave-uniform branch: EXEC stays all-1s

    const int  slice   = unit >> 4;          // (b*T + t)
    const int  tile    = unit & 15;          // 16-channel tile: c0 = tile*16
    const long in_base = (long)slice * (21 * 256) + tile * 16 + col;
    const long out_base= (long)slice * (10 * 256) + tile * 16 + col;

    v8f acc = {};

    // D(16x16) = A(16x24 = padded P) * B(24x16 = X rows, channel tile)
    // K split into 6 steps of 4. A built from constants; B loaded b32/lane:
    //   V0: lanes 0-15 -> row 4k+0, lanes 16-31 -> row 4k+2
    //   V1: lanes 0-15 -> row 4k+1, lanes 16-31 -> row 4k+3
    // Rows >= 21 clamped to 20: their A column is exactly 0, so the clamped
    // (in-bounds) data contributes nothing — no divergence, no OOB reads.
#pragma unroll
    for (int k = 0; k < 6; ++k) {
        const int n0 = 4 * k + 2 * half;      // K index held in reg 0
        const int n1 = 4 * k + 1 + 2 * half;  // K index held in reg 1

        v2f a;
        a.x = half ? pool_w(col, 4 * k + 2) : pool_w(col, 4 * k + 0);
        a.y = half ? pool_w(col, 4 * k + 3) : pool_w(col, 4 * k + 1);

        const int r0 = (n0 > 20) ? 20 : n0;
        const int r1 = (n1 > 20) ? 20 : n1;
        v2f b;
        b.x = in[in_base + (long)r0 * 256];
        b.y = in[in_base + (long)r1 * 256];

        // (neg_a, A, neg_b, B, c_mod, C, reuse_a, reuse_b)
        acc = __builtin_amdgcn_wmma_f32_16x16x4_f32(
            false, a, false, b, (short)0, acc, false, false);
    }

    // D layout: VGPR r -> row r (lanes 0-15) and row r+8 (lanes 16-31).
    // Keep rows 0..9 only.
#pragma unroll
    for (int r = 0; r < 8; ++r) {
        const int mrow = r + half * 8;
        if (mrow < 10) {
            out[out_base + (long)mrow * 256] = acc[r];
        }
    }
}

extern "C" void kernel_launch(void* const* d_in, const int* in_sizes, int n_in,
                              void* d_out, int out_size, void* d_ws, size_t ws_size,
                              hipStream_t stream) {
    const float* in = (const float*)d_in[0];
    float* out = (float*)d_out;

    const int slices  = in_sizes[0] / (21 * 256);  // B*T = 16384
    const int n_units = slices * 16;               // 16 channel tiles each

    const int waves_per_block = 8;                 // 256 threads
    const int blocks = (n_units + waves_per_block - 1) / waves_per_block;

    S_Down_Sampling_pool_wmma<<<blocks, 256, 0, stream>>>(in, out, n_units);
}